// ParallelBrainWithAdaptiveInhibition_53901839565358
// MI455X (gfx1250) — compile-verified
//
#include <hip/hip_runtime.h>
#include <stdint.h>

// ---------------------------------------------------------------------------
// ParallelBrainWithAdaptiveInhibition — CDNA5 (gfx1250)
// Memory-bound scatter-add; no matrix structure -> no WMMA. CDNA5 features:
//   * global_load_async_to_lds_b128 (+ s_wait_asynccnt) staging of the
//     per-neuron weight/index rows, tagged non-temporal (400MB stream > L2)
//   * global_atomic_add_f32 scatter from LDS
// Hot state (pot/th/post/fired ~1.6MB) stays cache-resident; streams are NT.
// ---------------------------------------------------------------------------

#define DECAY 0.95f
#define MAX_K 512   // LDS staging capacity (reference uses K=500)

// ---- CDNA5 async global->LDS copy (16 bytes per lane), non-temporal --------
// vdst = 32-bit LDS byte offset VGPR, vaddr = 64-bit global address VGPR pair.
__device__ __forceinline__ void async_copy16_nt(const void* gsrc, void* ldst) {
  unsigned lofs = (unsigned)(uintptr_t)ldst;   // LDS byte offset (addr[31:0])
  asm volatile("global_load_async_to_lds_b128 %0, %1, off th:TH_LOAD_NT"
               :: "v"(lofs), "v"(gsrc)
               : "memory");
}

__device__ __forceinline__ void wait_async0() {
#if __has_builtin(__builtin_amdgcn_s_wait_asynccnt)
  __builtin_amdgcn_s_wait_asynccnt(0);
#else
  asm volatile("s_wait_asynccnt 0" ::: "memory");
#endif
}

// ---- init: pot = potentials0, th = thresholds0 -----------------------------
__global__ void brain_init(float* __restrict__ pot, float* __restrict__ th,
                           const float* __restrict__ pot0,
                           const float* __restrict__ th0, int n) {
  int i = blockIdx.x * blockDim.x + threadIdx.x;
  if (i >= n) return;
  pot[i] = pot0[i];
  th[i]  = th0[i];
}

// ---- fused per-step elementwise -------------------------------------------
// pot += sign*post   (previous step's scatter; skipped at t==0)
// pot  = pot*DECAY + noise (+ext at t==0); fire; adapt threshold; post = 0
__global__ void brain_update(float* __restrict__ pot, float* __restrict__ th,
                             int* __restrict__ fired, float* __restrict__ post,
                             const float* __restrict__ noise_t,
                             const float* __restrict__ ext,
                             const float* __restrict__ inhib,
                             int n, int first_step) {
  int i = blockIdx.x * blockDim.x + threadIdx.x;
  if (i >= n) return;
  float p = pot[i];
  if (first_step) {
    p = p * DECAY + __builtin_nontemporal_load(noise_t + i) + ext[i];
  } else {
    float sign = 1.0f - 2.0f * inhib[i];
    p += sign * post[i];                       // apply previous scatter
    p = p * DECAY + __builtin_nontemporal_load(noise_t + i);
  }
  float t = th[i];
  int f = (p >= t) ? 1 : 0;
  t = t + (f ? 0.5f : 0.0f) - 0.1f;
  t = fminf(fmaxf(t, 1.0f), 100.0f);
  pot[i]   = p;
  th[i]    = t;
  fired[i] = f;
  post[i]  = 0.0f;   // re-zero accumulator for this step's scatter
}

// ---- scatter: one block per neuron; async-stage row into LDS, then atomics -
__global__ void __launch_bounds__(512)
brain_scatter(const int* __restrict__ fired,
              const float* __restrict__ weights,
              const int* __restrict__ conn,
              float* __restrict__ post, int k) {
  __shared__ __align__(16) float w_sh[MAX_K];
  __shared__ __align__(16) int   c_sh[MAX_K];

  const int neuron = blockIdx.x;
  if (fired[neuron] == 0) return;           // uniform branch: whole block exits

  const int t      = threadIdx.x;
  const int chunks = (k * 4) / 16;          // 16B chunks per row (k=500 -> 125)
  const float* wrow = weights + (size_t)neuron * (size_t)k;
  const int*   crow = conn    + (size_t)neuron * (size_t)k;

  // waves 0..3 stage the weight row, waves 8..11 stage the index row
  if (t < chunks) {
    async_copy16_nt(wrow + t * 4, &w_sh[t * 4]);
  } else if (t >= 256 && t < 256 + chunks) {
    int c = t - 256;
    async_copy16_nt(crow + c * 4, &c_sh[c * 4]);
  }
  wait_async0();        // each wave waits for its own ASYNCcnt
  __syncthreads();      // then all LDS writes in the block are visible

  if (t < k) {
    atomicAdd(&post[c_sh[t]], w_sh[t]);     // global_atomic_add_f32
  }
}

// ---- final apply after the last scatter: pot += (1 - 2*inhib) * post -------
__global__ void brain_apply(float* __restrict__ pot,
                            const float* __restrict__ post,
                            const float* __restrict__ inhib, int n) {
  int i = blockIdx.x * blockDim.x + threadIdx.x;
  if (i >= n) return;
  float sign = 1.0f - 2.0f * inhib[i];
  pot[i] += sign * post[i];
}

// ---------------------------------------------------------------------------
extern "C" void kernel_launch(void* const* d_in, const int* in_sizes, int n_in,
                              void* d_out, int out_size, void* d_ws, size_t ws_size,
                              hipStream_t stream) {
  const float* ext     = (const float*)d_in[0];   // [N]
  const float* weights = (const float*)d_in[1];   // [N,K]
  const int*   conn    = (const int*)  d_in[2];   // [N,K]
  const float* inhib   = (const float*)d_in[3];   // [N]
  const float* noise   = (const float*)d_in[4];   // [STEPS,N]
  const float* pot0    = (const float*)d_in[5];   // [N]
  const float* th0     = (const float*)d_in[6];   // [N]

  const int n     = in_sizes[0];
  const int k     = in_sizes[1] / n;              // 500
  const int steps = in_sizes[4] / n;              // 10

  float* pot   = (float*)d_out;                   // evolves in place
  float* th    = (float*)d_ws;                    // [N]
  float* post  = th + n;                          // [N]
  int*   fired = (int*)(post + n);                // [N]

  const int eb = 256;
  const int eg = (n + eb - 1) / eb;

  brain_init<<<eg, eb, 0, stream>>>(pot, th, pot0, th0, n);

  for (int t = 0; t < steps; ++t) {
    brain_update<<<eg, eb, 0, stream>>>(pot, th, fired, post,
                                        noise + (size_t)t * (size_t)n, ext,
                                        inhib, n, t == 0 ? 1 : 0);
    brain_scatter<<<n, 512, 0, stream>>>(fired, weights, conn, post, k);
  }
  brain_apply<<<eg, eb, 0, stream>>>(pot, post, inhib, n);
}